// Autoencoder_78288663872074
// MI455X (gfx1250) — compile-verified
//
#include <hip/hip_runtime.h>

// ---------------------------------------------------------------------------
// Autoencoder: 32 tiny Dense(ELU) layers over 4,194,304 rows of 5 floats.
// One batch row per SIMD lane slot; 4 rows per thread held as 2x float2 so
// the FMA chains lower to V_PK_FMA_F32 (VOP3P packed f32). Weights are
// wave-uniform scalar loads (s_load -> SGPR, verified in round 2). I/O staged
// through LDS with gfx1250 async-to-LDS ops (verified in round 2).
// ---------------------------------------------------------------------------

namespace {

constexpr int NL = 32;
// DIMS[0] = input width; DIMS[l] = output width of layer l (1-based layers).
constexpr int DIMS[NL + 1] = {
    5,
    // encoder: [5]*13, 4, 3, 2
    5, 5, 5, 5, 5, 5, 5, 5, 5, 5, 5, 5, 5, 4, 3, 2,
    // decoder: 5, 3, 4, [5]*10, 4, 3, 5
    5, 3, 4, 5, 5, 5, 5, 5, 5, 5, 5, 5, 5, 4, 3, 5};

__host__ __device__ constexpr int param_off(int l) {
  int o = 0;
  for (int k = 0; k < l; ++k) o += DIMS[k] * DIMS[k + 1] + DIMS[k + 1];
  return o;
}
constexpr int TOTAL_PARAMS = param_off(NL);  // 840 floats = 3360 bytes

constexpr int BLOCK = 256;
constexpr int RPT   = 4;              // rows per thread in the tiled kernel
constexpr int PAIRS = RPT / 2;        // float2 row-pairs per thread
constexpr int TROWS = BLOCK * RPT;    // 1024 rows per block
constexpr int TFLT  = TROWS * 5;      // 5120 floats per tile
constexpr int TVEC4 = TFLT / 4;       // 1280 float4 per tile

typedef float v2f __attribute__((ext_vector_type(2)));

// ------------------------------- ELU ---------------------------------------
__device__ __forceinline__ float elu1(float x) {
#if __has_builtin(__builtin_amdgcn_exp2f)
  float e = __builtin_amdgcn_exp2f(x * 1.44269504088896340736f) - 1.0f;
#else
  float e = exp2f(x * 1.44269504088896340736f) - 1.0f;
#endif
  return x > 0.0f ? x : e;
}

__device__ __forceinline__ v2f elu2(v2f x) {
  v2f r;
  r.x = elu1(x.x);
  r.y = elu1(x.y);
  return r;
}

// ---------------- packed (2 rows per v2f) dense layer -----------------------
template <int L>
__device__ __forceinline__ void dense_layer_pk(v2f (&h)[PAIRS][5],
                                               const float* __restrict__ P) {
  constexpr int IN  = DIMS[L];
  constexpr int OUT = DIMS[L + 1];
  constexpr int WO  = param_off(L);
  constexpr int BO  = WO + IN * OUT;
  v2f o[PAIRS][5];
#pragma unroll
  for (int j = 0; j < OUT; ++j) {
    const float b = P[BO + j];  // wave-uniform s_load
    v2f acc[PAIRS];
#pragma unroll
    for (int p = 0; p < PAIRS; ++p) acc[p] = (v2f){b, b};
#pragma unroll
    for (int i = 0; i < IN; ++i) {
      const float w = P[WO + i * OUT + j];  // wave-uniform s_load
      const v2f wv = (v2f){w, w};
#pragma unroll
      for (int p = 0; p < PAIRS; ++p)
        acc[p] = __builtin_elementwise_fma(h[p][i], wv, acc[p]);
    }
#pragma unroll
    for (int p = 0; p < PAIRS; ++p) o[p][j] = elu2(acc[p]);
  }
#pragma unroll
  for (int p = 0; p < PAIRS; ++p)
#pragma unroll
    for (int j = 0; j < OUT; ++j) h[p][j] = o[p][j];
}

template <int L>
__device__ __forceinline__ void run_layers_pk(v2f (&h)[PAIRS][5],
                                              const float* __restrict__ P) {
  if constexpr (L < NL) {
    dense_layer_pk<L>(h, P);
    run_layers_pk<L + 1>(h, P);
  }
}

// ---------------------- scalar layer (tail kernel) --------------------------
template <int L>
__device__ __forceinline__ void dense_layer_s(float (&h)[5],
                                              const float* __restrict__ P) {
  constexpr int IN  = DIMS[L];
  constexpr int OUT = DIMS[L + 1];
  constexpr int WO  = param_off(L);
  constexpr int BO  = WO + IN * OUT;
  float o[5];
#pragma unroll
  for (int j = 0; j < OUT; ++j) {
    float acc = P[BO + j];
#pragma unroll
    for (int i = 0; i < IN; ++i) acc = fmaf(h[i], P[WO + i * OUT + j], acc);
    o[j] = elu1(acc);
  }
#pragma unroll
  for (int j = 0; j < OUT; ++j) h[j] = o[j];
}

template <int L>
__device__ __forceinline__ void run_layers_s(float (&h)[5],
                                             const float* __restrict__ P) {
  if constexpr (L < NL) {
    dense_layer_s<L>(h, P);
    run_layers_s<L + 1>(h, P);
  }
}

// ---------------- gfx1250 async LDS path (guarded) --------------------------
#if __has_builtin(__builtin_amdgcn_global_load_async_to_lds_b128) && \
    __has_builtin(__builtin_amdgcn_global_store_async_from_lds_b128) && \
    __has_builtin(__builtin_amdgcn_s_wait_asynccnt)
#define USE_ASYNC_LDS 1
#else
#define USE_ASYNC_LDS 0
#endif

typedef int v4i __attribute__((vector_size(4 * sizeof(int))));
typedef __attribute__((address_space(1))) v4i* gv4i_ptr;  // global (__device__)
typedef __attribute__((address_space(3))) v4i* lv4i_ptr;  // LDS (__shared__)

#define AE_GLOBAL_V4(p) ((gv4i_ptr)(p))
#define AE_LDS_V4(p)    ((lv4i_ptr)(p))

// ------------------------------ main kernel ---------------------------------
__global__ __launch_bounds__(BLOCK) void ae_tiled_kernel(
    const float* __restrict__ x, float* __restrict__ out,
    const float* __restrict__ P) {
  __shared__ float4 tile4[TVEC4];
  float* tile = (float*)tile4;

  const int tid = threadIdx.x;
  const long long gbase = (long long)blockIdx.x * TFLT;
  const float4* __restrict__ xg = (const float4*)(x + gbase);
  float4* __restrict__ og = (float4*)(out + gbase);

  // ---- stage input tile into LDS (async on CDNA5) ----
#pragma unroll
  for (int k = 0; k < TVEC4 / BLOCK; ++k) {
    const int idx = k * BLOCK + tid;
#if USE_ASYNC_LDS
    __builtin_amdgcn_global_load_async_to_lds_b128(
        AE_GLOBAL_V4(xg + idx), AE_LDS_V4(tile4 + idx), 0, 0);
#else
    tile4[idx] = xg[idx];
#endif
  }
#if USE_ASYNC_LDS
  __builtin_amdgcn_s_wait_asynccnt(0);
#endif
  __syncthreads();

  // ---- pull my 4 rows as 2 row-pairs ----
  v2f h[PAIRS][5];
  const float* my = tile + tid * (RPT * 5);
#pragma unroll
  for (int p = 0; p < PAIRS; ++p)
#pragma unroll
    for (int j = 0; j < 5; ++j)
      h[p][j] = (v2f){my[(2 * p) * 5 + j], my[(2 * p + 1) * 5 + j]};

  // ---- 32 dense+ELU layers: packed f32 FMA, weights from SGPRs ----
  run_layers_pk<0>(h, P);

  // ---- write my rows back to the same LDS slots ----
  float* myw = tile + tid * (RPT * 5);
#pragma unroll
  for (int p = 0; p < PAIRS; ++p)
#pragma unroll
    for (int j = 0; j < 5; ++j) {
      myw[(2 * p) * 5 + j] = h[p][j].x;
      myw[(2 * p + 1) * 5 + j] = h[p][j].y;
    }
  __syncthreads();

  // ---- drain tile to global (async from LDS on CDNA5) ----
#pragma unroll
  for (int k = 0; k < TVEC4 / BLOCK; ++k) {
    const int idx = k * BLOCK + tid;
#if USE_ASYNC_LDS
    __builtin_amdgcn_global_store_async_from_lds_b128(
        AE_GLOBAL_V4(og + idx), AE_LDS_V4(tile4 + idx), 0, 0);
#else
    og[idx] = tile4[idx];
#endif
  }
#if USE_ASYNC_LDS
  __builtin_amdgcn_s_wait_asynccnt(0);
#endif
}

// ------------------------- tail kernel (1 row / thread) ---------------------
__global__ __launch_bounds__(BLOCK) void ae_tail_kernel(
    const float* __restrict__ x, float* __restrict__ out,
    const float* __restrict__ P, int row_begin, int row_end) {
  const int row = row_begin + blockIdx.x * BLOCK + threadIdx.x;
  if (row >= row_end) return;
  float h[5];
#pragma unroll
  for (int j = 0; j < 5; ++j) h[j] = x[(long long)row * 5 + j];
  run_layers_s<0>(h, P);
#pragma unroll
  for (int j = 0; j < 5; ++j) out[(long long)row * 5 + j] = h[j];
}

}  // namespace

// ------------------------------- launcher -----------------------------------
extern "C" void kernel_launch(void* const* d_in, const int* in_sizes, int n_in,
                              void* d_out, int out_size, void* d_ws,
                              size_t ws_size, hipStream_t stream) {
  (void)n_in;
  (void)out_size;
  (void)ws_size;

  // Pack all 32 layers' (W, b) into d_ws: d_in[0]=x, then W0,b0,W1,b1,...
  float* P = (float*)d_ws;
  int off = 0;
  for (int l = 0; l < NL; ++l) {
    const int din = DIMS[l], dout = DIMS[l + 1];
    (void)hipMemcpyAsync(P + off, d_in[1 + 2 * l],
                         (size_t)din * dout * sizeof(float),
                         hipMemcpyDeviceToDevice, stream);
    off += din * dout;
    (void)hipMemcpyAsync(P + off, d_in[2 + 2 * l],
                         (size_t)dout * sizeof(float),
                         hipMemcpyDeviceToDevice, stream);
    off += dout;
  }
  // off == TOTAL_PARAMS (840)

  const float* x = (const float*)d_in[0];
  float* out = (float*)d_out;
  const int rows = in_sizes[0] / 5;  // 4,194,304
  const int full_blocks = rows / TROWS;
  const int tail_begin = full_blocks * TROWS;

  if (full_blocks > 0) {
    ae_tiled_kernel<<<full_blocks, BLOCK, 0, stream>>>(x, out, P);
  }
  if (tail_begin < rows) {
    const int tail_rows = rows - tail_begin;
    const int tb = (tail_rows + BLOCK - 1) / BLOCK;
    ae_tail_kernel<<<tb, BLOCK, 0, stream>>>(x, out, P, tail_begin, rows);
  }
}